// LearnableUpsampler_37615323578437
// MI455X (gfx1250) — compile-verified
//
#include <hip/hip_runtime.h>
#include <math.h>

typedef __attribute__((ext_vector_type(2))) float v2f;
typedef __attribute__((ext_vector_type(8))) float v8f;

#define NB 32
#define NT 128
#define NH 256
#define NM 1000
#define NQ 4
#define NP 2

// ---------------- WMMA f32 16x16x4 helpers ----------------
// A (16x4, MxK): lanes 0-15 -> M=lane, K={k0,k0+1}; lanes 16-31 -> M=lane-16, K={k0+2,k0+3}
__device__ __forceinline__ v2f load_a_frag(const float* __restrict__ A, int lda, int row0, int k0, int lane) {
    int r  = row0 + (lane & 15);
    int kk = k0 + ((lane >> 4) << 1);
    v2f a; a[0] = A[r * lda + kk]; a[1] = A[r * lda + kk + 1];
    return a;
}
// B (4x16, KxN): lanes 0-15 -> N=lane, K={k0,k0+1}; lanes 16-31 -> N=lane-16, K={k0+2,k0+3}
__device__ __forceinline__ v2f load_b_frag(const float* __restrict__ B, int ldb, int col0, int k0, int lane) {
    int n  = col0 + (lane & 15);
    int kk = k0 + ((lane >> 4) << 1);
    v2f b; b[0] = B[kk * ldb + n]; b[1] = B[(kk + 1) * ldb + n];
    return b;
}
__device__ __forceinline__ v8f wmma4(v2f a, v2f b, v8f c) {
    return __builtin_amdgcn_wmma_f32_16x16x4_f32(false, a, false, b, (short)0, c, false, false);
}
__device__ __forceinline__ float silu(float x) { return x / (1.0f + expf(-x)); }

// ---------------- Kernel 1: h = x @ proj_w + proj_b  (4096x256 @ 256x256) ----------------
__global__ void k_proj(const float* __restrict__ x, const float* __restrict__ pw,
                       const float* __restrict__ pb, float* __restrict__ h) {
    int lane = threadIdx.x;
    int col0 = blockIdx.x * 16;   // 16 n-tiles
    int row0 = blockIdx.y * 16;   // 256 m-tiles (B*T=4096 rows)
    v8f acc = {};
    for (int k0 = 0; k0 < NH; k0 += 4)
        acc = wmma4(load_a_frag(x, NH, row0, k0, lane), load_b_frag(pw, NH, col0, k0, lane), acc);
    int n = col0 + (lane & 15);
    int rbase = row0 + ((lane >> 4) << 3);
    float bias = pb[n];
#pragma unroll
    for (int r = 0; r < 8; r++) h[(rbase + r) * NH + n] = acc[r] + bias;
}

// ---------------- Kernel 2: conv1d(K=3,SAME) on masked h + LN(8) + SiLU * valid ----------------
__global__ void k_conv(const float* __restrict__ h, const int* __restrict__ xlen,
                       const float* __restrict__ cw, const float* __restrict__ cb,
                       const float* __restrict__ cg, const float* __restrict__ cbe,
                       float* __restrict__ hc) {
    int b = blockIdx.x, t = threadIdx.x;
    int len = xlen[b];
    float o8[8];
#pragma unroll
    for (int j = 0; j < 8; j++) o8[j] = 0.0f;
    for (int k = 0; k < 3; k++) {
        int tt = t + k - 1;
        if (tt < 0 || tt >= NT || tt >= len) continue;  // SAME pad + valid mask on conv input
        const float* hp = h + (b * NT + tt) * NH;
        for (int i = 0; i < NH; i++) {
            float hv = hp[i];
#pragma unroll
            for (int j = 0; j < 8; j++) o8[j] += hv * cw[(j * NH + i) * 3 + k];
        }
    }
    float v[8], mu = 0.0f;
#pragma unroll
    for (int j = 0; j < 8; j++) { v[j] = o8[j] + cb[j]; mu += v[j]; }
    mu *= 0.125f;
    float var = 0.0f;
#pragma unroll
    for (int j = 0; j < 8; j++) { float d = v[j] - mu; var += d * d; }
    var *= 0.125f;
    float rs = rsqrtf(var + 1e-5f);
    float valid = (t < len) ? 1.0f : 0.0f;
#pragma unroll
    for (int j = 0; j < 8; j++) {
        float y = (v[j] - mu) * rs * cg[j] + cbe[j];
        hc[(b * NT + t) * 8 + j] = silu(y) * valid;
    }
}

// ---------------- Kernel 3: duration cumsum -> s, e, mel_lens ----------------
__global__ void k_durs(const float* __restrict__ dur, float* __restrict__ sArr,
                       float* __restrict__ eArr, int* __restrict__ lens,
                       float* __restrict__ out_lens) {
    int b = threadIdx.x;
    if (b >= NB) return;
    float c = 0.0f;
    for (int t = 0; t < NT; t++) {
        float d = dur[b * NT + t];
        c += d;
        eArr[b * NT + t] = c;
        sArr[b * NT + t] = c - d;
    }
    int L = (int)rintf(c);
    if (L > NM) L = NM;
    lens[b] = L;
    out_lens[b] = (float)L;
}

// ---------------- Kernel 4: mel_mask output (B,M,T) ----------------
__global__ void k_mask(const int* __restrict__ lens, float* __restrict__ mout) {
    int idx = blockIdx.x * blockDim.x + threadIdx.x;
    if (idx >= NB * NM * NT) return;
    int bm = idx / NT;
    int m = bm % NM, b = bm / NM;
    mout[idx] = (m >= lens[b]) ? 1.0f : 0.0f;
}

// ---------------- Kernel 5: mlp_q softmax -> w  +  mlp_p -> wc8 ----------------
// one block (128 threads = 4 waves) per (b,m); thread = phone t
__global__ void k_qp(const float* __restrict__ hc, const float* __restrict__ sArr,
                     const float* __restrict__ eArr, const int* __restrict__ xlen,
                     const int* __restrict__ lens,
                     const float* __restrict__ qw1, const float* __restrict__ qb1,
                     const float* __restrict__ qg, const float* __restrict__ qbe,
                     const float* __restrict__ qw2, const float* __restrict__ qb2,
                     const float* __restrict__ pw1, const float* __restrict__ pb1,
                     const float* __restrict__ pg, const float* __restrict__ pbe,
                     const float* __restrict__ pw2, const float* __restrict__ pb2,
                     float* __restrict__ wout, float* __restrict__ wc8) {
    __shared__ float red[4][4];
    __shared__ float red8[4][8];
    int bm = blockIdx.x;
    int b = bm / NM, m = bm % NM;
    int t = threadIdx.x;
    int wave = t >> 5, lane = t & 31;
    bool xm = (t >= xlen[b]);
    bool mpad = (m >= lens[b]);
    bool msk = xm || mpad;
    float tt = (float)(m + 1);
    float Ev = msk ? 0.0f : (eArr[b * NT + t] - tt);
    float Sv = msk ? 0.0f : (tt - sArr[b * NT + t]);
    float f[10];
#pragma unroll
    for (int j = 0; j < 8; j++) f[j] = hc[(b * NT + t) * 8 + j];
    f[8] = Sv; f[9] = Ev;

    // mlp_q: 10->4, LN(4), silu, 4->4
    float a4[4];
#pragma unroll
    for (int j = 0; j < 4; j++) {
        float s = qb1[j];
        for (int i = 0; i < 10; i++) s += f[i] * qw1[i * 4 + j];
        a4[j] = s;
    }
    float mu = 0.25f * (a4[0] + a4[1] + a4[2] + a4[3]);
    float var = 0.0f;
#pragma unroll
    for (int j = 0; j < 4; j++) { float d = a4[j] - mu; var += d * d; }
    var *= 0.25f;
    float rs = rsqrtf(var + 1e-5f);
    float u4[4];
#pragma unroll
    for (int j = 0; j < 4; j++) u4[j] = silu((a4[j] - mu) * rs * qg[j] + qbe[j]);
    float sc[4];
#pragma unroll
    for (int j = 0; j < 4; j++) {
        float s = qb2[j];
        for (int i = 0; i < 4; i++) s += u4[i] * qw2[i * 4 + j];
        sc[j] = xm ? -INFINITY : s;
    }

    // masked softmax over t (128 threads) per q: max then sum
    float gmax[4];
#pragma unroll
    for (int j = 0; j < 4; j++) {
        float v = sc[j];
        for (int off = 16; off > 0; off >>= 1) v = fmaxf(v, __shfl_xor(v, off, 32));
        if (lane == 0) red[wave][j] = v;
    }
    __syncthreads();
#pragma unroll
    for (int j = 0; j < 4; j++)
        gmax[j] = fmaxf(fmaxf(red[0][j], red[1][j]), fmaxf(red[2][j], red[3][j]));
    __syncthreads();
    float ex[4], gsum[4];
#pragma unroll
    for (int j = 0; j < 4; j++) {
        float v = expf(sc[j] - gmax[j]);  // -inf -> 0 for padded phones
        ex[j] = v;
        for (int off = 16; off > 0; off >>= 1) v += __shfl_xor(v, off, 32);
        if (lane == 0) red[wave][j] = v;
    }
    __syncthreads();
#pragma unroll
    for (int j = 0; j < 4; j++)
        gsum[j] = red[0][j] + red[1][j] + red[2][j] + red[3][j];
    float w4[4];
#pragma unroll
    for (int j = 0; j < 4; j++) {
        w4[j] = mpad ? 0.0f : (ex[j] / gsum[j]);
        // store in (b,q,m,t) layout: output layout AND contiguous GEMM-A layout
        wout[(((long)b * NQ + j) * NM + m) * NT + t] = w4[j];
    }

    // mlp_p: 10->2, LN(2), silu, 2->2
    float a2[2];
#pragma unroll
    for (int p = 0; p < 2; p++) {
        float s = pb1[p];
        for (int i = 0; i < 10; i++) s += f[i] * pw1[i * 2 + p];
        a2[p] = s;
    }
    float mu2 = 0.5f * (a2[0] + a2[1]);
    float d0 = a2[0] - mu2, d1 = a2[1] - mu2;
    float rs2 = rsqrtf(0.5f * (d0 * d0 + d1 * d1) + 1e-5f);
    float u0 = silu(d0 * rs2 * pg[0] + pbe[0]);
    float u1 = silu(d1 * rs2 * pg[1] + pbe[1]);
    float c2[2];
#pragma unroll
    for (int p = 0; p < 2; p++) c2[p] = pb2[p] + u0 * pw2[0 * 2 + p] + u1 * pw2[1 * 2 + p];

    // wc8[b,m,q*2+p] = sum_t w[t,q]*c[t,p]
#pragma unroll
    for (int idx = 0; idx < 8; idx++) {
        int q = idx >> 1, p = idx & 1;
        float v = w4[q] * c2[p];
        for (int off = 16; off > 0; off >>= 1) v += __shfl_xor(v, off, 32);
        if (lane == 0) red8[wave][idx] = v;
    }
    __syncthreads();
    if (t < 8) wc8[(long)bm * 8 + t] = red8[0][t] + red8[1][t] + red8[2][t] + red8[3][t];
}

// ---------------- Kernel 6: wh einsum as GEMM per batch: (4000x128)@(128x256) ----------------
__global__ void k_wh(const float* __restrict__ wout, const float* __restrict__ x,
                     float* __restrict__ whq) {
    int lane = threadIdx.x;
    int col0 = blockIdx.x * 16;   // 16 n-tiles over H=256
    int row0 = blockIdx.y * 16;   // 250 m-tiles over Q*M=4000
    int b = blockIdx.z;
    const float* A = wout + (long)b * 4000 * NT;
    const float* B = x + (long)b * NT * NH;
    float* C = whq + (long)b * 4000 * NH;
    v8f acc = {};
    for (int k0 = 0; k0 < NT; k0 += 4)
        acc = wmma4(load_a_frag(A, NT, row0, k0, lane), load_b_frag(B, NH, col0, k0, lane), acc);
    int n = col0 + (lane & 15);
    int rbase = row0 + ((lane >> 4) << 3);
#pragma unroll
    for (int r = 0; r < 8; r++) C[(rbase + r) * NH + n] = acc[r];
}

// ---------------- Kernel 7: fused proj1 + wc8@proj2 + mask -> LDS -> projo ----------------
// 128 threads (4 waves), 16 rows of (b,m) per block; 2000 blocks
__global__ void k_out(const float* __restrict__ whq, const float* __restrict__ wc8,
                      const float* __restrict__ p1w, const float* __restrict__ p1b,
                      const float* __restrict__ p2w, const float* __restrict__ p2b,
                      const float* __restrict__ pow_, const float* __restrict__ pob,
                      const int* __restrict__ lens, float* __restrict__ oout) {
    __shared__ float tmp[16][NH + 4];
    int tid = threadIdx.x;
    int wave = tid >> 5, lane = tid & 31;
    int R0 = blockIdx.x * 16;  // global row block over B*M = 32000

    // stage 1: tmp(16x256) = whq_rows(16x1024) @ proj1_w + wc8 @ proj2_w + biases, mel-pad masked
    int rA = R0 + (lane & 15);
    int bA = rA / NM, mA = rA % NM;
    long baseA = ((long)bA * 4000 + mA) * NH;  // + q*NM*NH + h gives A[(b,m), q*256+h]
    for (int nt = wave * 4; nt < wave * 4 + 4; nt++) {
        int col0 = nt * 16;
        int n = col0 + (lane & 15);
        v8f acc = {};
        for (int k0 = 0; k0 < 1024; k0 += 4) {
            int kk = k0 + ((lane >> 4) << 1);
            int q = kk >> 8, hh = kk & 255;
            const float* pA = whq + baseA + (long)q * NM * NH + hh;
            v2f a; a[0] = pA[0]; a[1] = pA[1];
            v2f bf; bf[0] = p1w[kk * NH + n]; bf[1] = p1w[(kk + 1) * NH + n];
            acc = wmma4(a, bf, acc);
        }
        int lrbase = ((lane >> 4) << 3);
        float bias = p1b[n] + p2b[n];
#pragma unroll
        for (int r = 0; r < 8; r++) {
            int lr = lrbase + r;
            int R = R0 + lr;
            int bR = R / NM, mR = R % NM;
            float v = acc[r] + bias;
            const float* wcp = wc8 + (long)R * 8;
#pragma unroll
            for (int j = 0; j < 8; j++) v += wcp[j] * p2w[j * NH + n];
            if (mR >= lens[bR]) v = 0.0f;  // where(mel_pad, 0, wh+wc)
            tmp[lr][n] = v;
        }
    }
    __syncthreads();

    // stage 2: o(16x512) = tmp(16x256) @ projo_w + projo_b
    for (int nt = wave * 8; nt < wave * 8 + 8; nt++) {
        int col0 = nt * 16;
        int n = col0 + (lane & 15);
        v8f acc = {};
        for (int k0 = 0; k0 < NH; k0 += 4) {
            int kk = k0 + ((lane >> 4) << 1);
            int r = lane & 15;
            v2f a; a[0] = tmp[r][kk]; a[1] = tmp[r][kk + 1];
            v2f bf; bf[0] = pow_[kk * 512 + n]; bf[1] = pow_[(kk + 1) * 512 + n];
            acc = wmma4(a, bf, acc);
        }
        int lrbase = ((lane >> 4) << 3);
        float bias = pob[n];
#pragma unroll
        for (int r = 0; r < 8; r++) {
            int R = R0 + lrbase + r;
            oout[(long)R * 512 + n] = acc[r] + bias;
        }
    }
}

// ---------------- host launch ----------------
extern "C" void kernel_launch(void* const* d_in, const int* in_sizes, int n_in,
                              void* d_out, int out_size, void* d_ws, size_t ws_size,
                              hipStream_t stream) {
    const float* x       = (const float*)d_in[0];
    // d_in[1] = x_mask (bool) -- derived from x_lengths instead
    const float* x_dur   = (const float*)d_in[2];
    const int*   x_len   = (const int*)d_in[3];
    const float* proj_w  = (const float*)d_in[4];
    const float* proj_b  = (const float*)d_in[5];
    const float* conv_w  = (const float*)d_in[6];
    const float* conv_b  = (const float*)d_in[7];
    const float* conv_g  = (const float*)d_in[8];
    const float* conv_be = (const float*)d_in[9];
    const float* q_w1    = (const float*)d_in[10];
    const float* q_b1    = (const float*)d_in[11];
    const float* q_g     = (const float*)d_in[12];
    const float* q_be    = (const float*)d_in[13];
    const float* q_w2    = (const float*)d_in[14];
    const float* q_b2    = (const float*)d_in[15];
    const float* p_w1    = (const float*)d_in[16];
    const float* p_b1    = (const float*)d_in[17];
    const float* p_g     = (const float*)d_in[18];
    const float* p_be    = (const float*)d_in[19];
    const float* p_w2    = (const float*)d_in[20];
    const float* p_b2    = (const float*)d_in[21];
    const float* p1w     = (const float*)d_in[22];
    const float* p1b     = (const float*)d_in[23];
    const float* p2w     = (const float*)d_in[24];
    const float* p2b     = (const float*)d_in[25];
    const float* pow_    = (const float*)d_in[26];
    const float* pob     = (const float*)d_in[27];

    float* out   = (float*)d_out;
    float* o_o    = out;                       // (B,M,2H)  16,384,000
    float* o_mask = out + 16384000;            // (B,M,T)    4,096,000
    float* o_lens = out + 20480000;            // (B,)              32
    float* o_w    = out + 20480032;            // (B,Q,M,T) 16,384,000

    float* ws   = (float*)d_ws;
    float* h    = ws;                          // B*T*H   = 1,048,576
    float* hc   = h + 1048576;                 // B*T*8   =    32,768
    float* sArr = hc + 32768;                  // B*T     =     4,096
    float* eArr = sArr + 4096;                 // B*T     =     4,096
    int*   lens = (int*)(eArr + 4096);         // B       =        32
    float* wc8  = (float*)(lens + 32);         // B*M*8   =   256,000
    float* whq  = wc8 + 256000;                // B*4000*H = 32,768,000

    k_proj<<<dim3(16, 256), 32, 0, stream>>>(x, proj_w, proj_b, h);
    k_conv<<<NB, NT, 0, stream>>>(h, x_len, conv_w, conv_b, conv_g, conv_be, hc);
    k_durs<<<1, 32, 0, stream>>>(x_dur, sArr, eArr, lens, o_lens);
    k_mask<<<(NB * NM * NT + 255) / 256, 256, 0, stream>>>(lens, o_mask);
    k_qp<<<NB * NM, NT, 0, stream>>>(hc, sArr, eArr, x_len, lens,
                                     q_w1, q_b1, q_g, q_be, q_w2, q_b2,
                                     p_w1, p_b1, p_g, p_be, p_w2, p_b2,
                                     o_w, wc8);
    k_wh<<<dim3(16, 250, NB), 32, 0, stream>>>(o_w, x, whq);
    k_out<<<2000, 128, 0, stream>>>(whq, wc8, p1w, p1b, p2w, p2b, pow_, pob, lens, o_o);
}